// MixtureMemoryModel_38585986187952
// MI455X (gfx1250) — compile-verified
//
#include <hip/hip_runtime.h>
#include <hip/hip_bf16.h>
#include <math.h>

typedef __attribute__((ext_vector_type(2))) float v2f;
typedef __attribute__((ext_vector_type(4))) float v4f;
typedef __attribute__((ext_vector_type(8))) float v8f;

#define DDIM   4096
#define MROWS  16384
#define DCHUNK 1024   // columns per block (4 chunks cover DDIM)
#define NSPLIT (DDIM / DCHUNK)

// ---------------------------------------------------------------------------
// Zero the sq_dist accumulator (workspace is poisoned; atomics need zeros).
// ---------------------------------------------------------------------------
__global__ void mm_zero_kernel(float* __restrict__ p, int n) {
    int i = blockIdx.x * blockDim.x + threadIdx.x;
    if (i < n) p[i] = 0.0f;
}

// ---------------------------------------------------------------------------
// Kernel 1: sq_dist[m] += sum_{k in chunk} (rep[k] - bank[m][k])^2
//
// One wave owns a 16-row tile over a 1024-column chunk. B = all-ones, so
// v_wmma_f32_16x16x4_f32 computes D[m,n] = sum_k A[m,k] — a 16-row x 4-K
// reduction per instruction. Because only the per-row SUM matters, the exact
// K-slot each value occupies is irrelevant; we exploit this to stream with
// global_load_b128: lane L (row L%16) covers columns c+4*(L/16) .. +3, so one
// b128 load (16B/lane, 32B contiguous per row, 512B/wave) feeds two WMMAs.
// ---------------------------------------------------------------------------
__global__ __launch_bounds__(256) void mm_sqdist_kernel(
        const float* __restrict__ rep,
        const float* __restrict__ bank,
        float* __restrict__ sqd) {
    __shared__ __align__(16) float srep[DCHUNK];

    const int tid  = threadIdx.x;
    const int wave = tid >> 5;
    const int lane = tid & 31;
    const int c0   = blockIdx.x * DCHUNK;

    // Stage this block's rep chunk into LDS (4 KB), shared by all 8 waves.
    for (int i = tid; i < DCHUNK; i += 256) srep[i] = rep[c0 + i];
    __syncthreads();

    const int tile = blockIdx.y * 8 + wave;      // 16-row tile index
    const int row  = (tile << 4) + (lane & 15);  // this lane's matrix row
    const int kq   = (lane >> 4) << 2;           // 0 for lanes 0-15, 4 for 16-31
    const float* rowp = bank + (size_t)row * DDIM + c0 + kq;
    const float* srp  = srep + kq;

    v8f acc = {};
    const v2f ones = {1.0f, 1.0f};

    for (int c = 0; c < DCHUNK; c += 64) {
        // Stream ahead ~1KB within the row (speculative; OOB is dropped).
        __builtin_prefetch(rowp + c + 256, 0, 1);
#pragma unroll
        for (int k = 0; k < 64; k += 8) {
            v4f mv = *(const v4f*)(rowp + c + k);   // 16B/lane, 16B-aligned
            v4f rv = *(const v4f*)(srp + c + k);    // LDS b128 broadcast read
            v4f d  = mv - rv;
            d = d * d;
            v2f a0 = {d.x, d.y};
            v2f a1 = {d.z, d.w};
            // D = A x ones + C : per-row sums of diff^2, 4 K-values per WMMA
            acc = __builtin_amdgcn_wmma_f32_16x16x4_f32(
                false, a0, false, ones, (short)0, acc, false, false);
            acc = __builtin_amdgcn_wmma_f32_16x16x4_f32(
                false, a1, false, ones, (short)0, acc, false, false);
        }
    }

    // C/D layout: lane L, VGPR j -> M = j + 8*(L>>4), N = L&15 (replicated).
    // Lanes with N==0 (lanes 0 and 16) flush rows [0..7] / [8..15].
    if ((lane & 15) == 0) {
        const int rbase = (tile << 4) + ((lane >> 4) << 3);
#pragma unroll
        for (int j = 0; j < 8; ++j)
            atomicAdd(&sqd[rbase + j], acc[j]);
    }
}

// ---------------------------------------------------------------------------
// Kernel 2: Gaussian log-pdfs, max, logsumexp, decision. One 1024-thread block.
// log_probs[i] = -0.5*(D*log(2*pi*var_i) + sq[i]/var_i), var_i = (M-i) + 1e-3
// ---------------------------------------------------------------------------
__global__ __launch_bounds__(1024) void mm_finalize_kernel(
        const float* __restrict__ sqd, float* __restrict__ out) {
    __shared__ float red[1024];
    const int tid = threadIdx.x;
    const float TWO_PI = 6.2831853071795864769f;

    // Pass 1: max of log_probs
    float lmax = -INFINITY;
    for (int i = tid; i < MROWS; i += 1024) {
        float var = (float)(MROWS - i) + 0.001f;     // NOISE_SLOPE=1, OFFSET=1e-3
        float lp  = -0.5f * (4096.0f * __logf(TWO_PI * var) + sqd[i] / var);
        lmax = fmaxf(lmax, lp);
    }
    red[tid] = lmax;
    __syncthreads();
    for (int s = 512; s > 0; s >>= 1) {
        if (tid < s) red[tid] = fmaxf(red[tid], red[tid + s]);
        __syncthreads();
    }
    const float gmax = red[0];
    __syncthreads();

    // Pass 2: sum exp(lp - gmax)  (recompute lp; sq_dist sits in L2)
    float lsum = 0.0f;
    for (int i = tid; i < MROWS; i += 1024) {
        float var = (float)(MROWS - i) + 0.001f;
        float lp  = -0.5f * (4096.0f * __logf(TWO_PI * var) + sqd[i] / var);
        lsum += __expf(lp - gmax);
    }
    red[tid] = lsum;
    __syncthreads();
    for (int s = 512; s > 0; s >>= 1) {
        if (tid < s) red[tid] += red[tid + s];
        __syncthreads();
    }

    if (tid == 0) {
        float ll  = gmax + __logf(red[0]) - __logf((float)MROWS);
        float thr = __logf(0.5f) + gmax;             // CRITERION = 0.5
        out[0] = (ll >= thr) ? 1.0f : 0.0f;          // decision
        out[1] = ll;                                  // log_likelihood
        out[2] = thr;                                 // threshold
    }
}

// ---------------------------------------------------------------------------
extern "C" void kernel_launch(void* const* d_in, const int* in_sizes, int n_in,
                              void* d_out, int out_size, void* d_ws, size_t ws_size,
                              hipStream_t stream) {
    (void)in_sizes; (void)n_in; (void)out_size; (void)ws_size;
    const float* rep  = (const float*)d_in[0];   // [4096]
    const float* bank = (const float*)d_in[1];   // [16384, 4096]
    float* out = (float*)d_out;                  // [3]: decision, ll, threshold
    float* sqd = (float*)d_ws;                   // [16384] partial sq distances

    mm_zero_kernel<<<MROWS / 256, 256, 0, stream>>>(sqd, MROWS);

    dim3 grid(NSPLIT, MROWS / 16 / 8);           // (4, 128), 256 thr = 8 waves
    mm_sqdist_kernel<<<grid, 256, 0, stream>>>(rep, bank, sqd);

    mm_finalize_kernel<<<1, 1024, 0, stream>>>(sqd, out);
}